// RNN_4398046511300
// MI455X (gfx1250) — compile-verified
//
#include <hip/hip_runtime.h>
#include <hip/hip_bf16.h>

typedef __attribute__((ext_vector_type(2))) float v2f;
typedef __attribute__((ext_vector_type(8))) float v8f;

#define HID 50

__device__ __forceinline__ float fast_tanh(float x) {
    float r;
    asm("v_tanh_f32 %0, %1" : "=v"(r) : "v"(x));
    return r;
}

// One wave (32 lanes) per block; each wave owns a 16-row batch tile and runs
// the whole T-step recurrence autonomously.
//
// Orientation: compute h^T so that WMMA D-layout (lane = batch col) matches the
// next step's B-layout (lane = batch col). LDS holds h^T in the exact per-lane
// B-operand order:
//   dword(K, n) = 64*(K>>2) + 32*((K>>1)&1) + (K&1) + 2*n
// so a B chunk kc is one float2 load per lane at [64*kc + 32*hi + 2*n].
__global__ __launch_bounds__(32) void rnn_wmma_f32_kernel(
    const float* __restrict__ x,     // [B, T]
    const float* __restrict__ W_ih,  // [H]
    const float* __restrict__ W_hh,  // [H, H]
    const float* __restrict__ b_ih,  // [H]
    const float* __restrict__ b_hh,  // [H]
    const float* __restrict__ W_fc,  // [H]
    const float* __restrict__ b_fc,  // [1]
    float* __restrict__ out,         // [B]
    int T) {
    __shared__ float hbuf[1024];  // 64 rows (K) x 16 cols (batch), B-operand order

    const int lane = threadIdx.x;
    const int n    = lane & 15;   // batch col within tile / M row within A
    const int hi   = lane >> 4;   // lane half
    const int batch_base = blockIdx.x * 16;

    // ---- A = W_hh, loop-invariant in VGPRs. A-layout (16x16x4 f32):
    //   lanes 0-15: M=0..15 ; lanes 16-31: M=0..15
    //   vgpr0: K = base + 2*hi ; vgpr1: K = base + 2*hi + 1
    v2f A[4][13];
#pragma unroll
    for (int mt = 0; mt < 4; ++mt) {
#pragma unroll
        for (int kc = 0; kc < 13; ++kc) {
            int i = 16 * mt + n;        // row of W_hh (output index)
            int j = 4 * kc + 2 * hi;    // col of W_hh (input index)
            float a0 = (i < HID && j     < HID) ? W_hh[i * HID + j]     : 0.0f;
            float a1 = (i < HID && j + 1 < HID) ? W_hh[i * HID + j + 1] : 0.0f;
            v2f av; av.x = a0; av.y = a1;
            A[mt][kc] = av;
        }
    }

    // ---- per-lane D-layout vectors of W_ih and (b_ih + b_hh): element (mt, r)
    // lives at i = 16*mt + 8*hi + r (zero-padded past H).
    float wih[4][8], bias[4][8];
#pragma unroll
    for (int mt = 0; mt < 4; ++mt) {
#pragma unroll
        for (int r = 0; r < 8; ++r) {
            int i = 16 * mt + 8 * hi + r;
            wih[mt][r]  = (i < HID) ? W_ih[i] : 0.0f;
            bias[mt][r] = (i < HID) ? (b_ih[i] + b_hh[i]) : 0.0f;
        }
    }

    // h0 = 0
#pragma unroll
    for (int k = 0; k < 32; ++k) hbuf[k * 32 + lane] = 0.0f;
    __syncthreads();

    const int sbase = 128 * hi + 2 * n;  // store base (D rows -> B layout)
    const int lbase = 32 * hi + 2 * n;   // load base  (B chunk float2)

    const float* xrow = x + (size_t)(batch_base + n) * (size_t)T;
    const int ngroups = T >> 2;  // T is a multiple of 4 (T=512)

    float4 xc = ((const float4*)xrow)[0];
    for (int g = 0; g < ngroups; ++g) {
        // prefetch next 4 timesteps of x while this group computes
        float4 xn = (g + 1 < ngroups) ? ((const float4*)xrow)[g + 1]
                                      : make_float4(0.f, 0.f, 0.f, 0.f);
        float xs[4] = {xc.x, xc.y, xc.z, xc.w};
#pragma unroll
        for (int u = 0; u < 4; ++u) {
            float xt = xs[u];

            // B operands: h_prev^T, 13 chunks of K=4
            v2f bv[13];
#pragma unroll
            for (int kc = 0; kc < 13; ++kc)
                bv[kc] = *(const v2f*)&hbuf[64 * kc + lbase];

#pragma unroll
            for (int mt = 0; mt < 4; ++mt) {
                // accumulator init: xproj^T + b_ih + b_hh
                v8f c;
#pragma unroll
                for (int r = 0; r < 8; ++r)
                    c[r] = fmaf(wih[mt][r], xt, bias[mt][r]);

                // D = W_hh(tile mt) * h_prev^T + c   (13 chained f32 WMMAs)
#pragma unroll
                for (int kc = 0; kc < 13; ++kc)
                    c = __builtin_amdgcn_wmma_f32_16x16x4_f32(
                        false, A[mt][kc], false, bv[kc], (short)0, c, false, false);

                // h_new^T = tanh(D), stored straight into B-operand layout
#pragma unroll
                for (int r = 0; r < 8; ++r) {
                    int coff = (r >> 2) * 64 + ((r >> 1) & 1) * 32 + (r & 1);
                    hbuf[sbase + 256 * mt + coff] = fast_tanh(c[r]);
                }
            }
        }
        xc = xn;
    }
    __syncthreads();

    // ---- epilogue: out[b] = sum_i h_T[i, b] * W_fc[i] + b_fc
    float partial = 0.0f;
#pragma unroll
    for (int ii = 0; ii < 25; ++ii) {
        int i  = hi * 25 + ii;
        int dw = 64 * (i >> 2) + 32 * ((i >> 1) & 1) + (i & 1) + 2 * n;
        partial = fmaf(hbuf[dw], W_fc[i], partial);
    }
    if (hi == 1) hbuf[n] = partial;
    __syncthreads();
    if (hi == 0) out[batch_base + n] = partial + hbuf[n] + b_fc[0];
}

extern "C" void kernel_launch(void* const* d_in, const int* in_sizes, int n_in,
                              void* d_out, int out_size, void* d_ws, size_t ws_size,
                              hipStream_t stream) {
    const float* x    = (const float*)d_in[0];
    const float* W_ih = (const float*)d_in[1];
    const float* W_hh = (const float*)d_in[2];
    const float* b_ih = (const float*)d_in[3];
    const float* b_hh = (const float*)d_in[4];
    const float* W_fc = (const float*)d_in[5];
    const float* b_fc = (const float*)d_in[6];
    float* out = (float*)d_out;

    const int T = 512;                 // reference shape (I = 1)
    const int B = in_sizes[0] / T;     // x is [B, T, 1]
    const int nblocks = B / 16;        // one 16-row batch tile per wave/block

    rnn_wmma_f32_kernel<<<nblocks, 32, 0, stream>>>(
        x, W_ih, W_hh, b_ih, b_hh, W_fc, b_fc, out, T);
}